// MFTF_43447889167057
// MI455X (gfx1250) — compile-verified
//
#include <hip/hip_runtime.h>
#include <hip/hip_bf16.h>
#include <math.h>

typedef __attribute__((ext_vector_type(2))) float v2f;
typedef __attribute__((ext_vector_type(8))) float v8f;

#define NPOS   51
#define NAGG   50
#define NNEG   200
#define NEASY  50
#define NHARD  50
#define DIM    128
#define NTILE  13   // ceil(200/16) tiles of 16 negatives

__device__ __forceinline__ float softplusf(float x) {
    // log(1 + exp(x)), numerically stable
    return log1pf(expf(-fabsf(x))) + fmaxf(x, 0.0f);
}

__global__ __launch_bounds__(256) void MFTF_main_kernel(
    const int*   __restrict__ x_pos,   // (256, 51)
    const int*   __restrict__ x_neg,   // (256, 200)
    const float* __restrict__ emb,     // (1000000, 128)
    float*       __restrict__ part_pos, // (256)
    float*       __restrict__ part_neg) // (256)
{
    __shared__ __align__(16) float s_wavesum[8][DIM];
    __shared__ __align__(16) float s_avg[DIM];
    __shared__ __align__(16) float s_negprod[NTILE * 16];
    __shared__ float s_posw[8];
    __shared__ float s_cntw[8];
    __shared__ float s_red[256];

    const int b    = blockIdx.x;
    const int tid  = threadIdx.x;
    const int wave = tid >> 5;
    const int lane = tid & 31;

    const int* xp = x_pos + b * NPOS;
    const int* xn = x_neg + b * NNEG;

    // ---------- Phase A: masked sum of X_agg rows + positive-pair dots ----------
    float4 acc = make_float4(0.f, 0.f, 0.f, 0.f);
    float pos_acc = 0.f, cnt_acc = 0.f;
    for (int t = wave; t < NAGG; t += 8) {
        const int i0 = xp[t];
        const int i1 = xp[t + 1];
        const float w = (i1 != 0) ? 1.0f : 0.0f;
        // full-wave 512B coalesced row loads (32 lanes x b128)
        const float4 a0 = ((const float4*)(emb + (size_t)i0 * DIM))[lane];
        const float4 a1 = ((const float4*)(emb + (size_t)i1 * DIM))[lane];
        acc.x += w * a0.x; acc.y += w * a0.y; acc.z += w * a0.z; acc.w += w * a0.w;
        float d = a0.x*a1.x + a0.y*a1.y + a0.z*a1.z + a0.w*a1.w;
        #pragma unroll
        for (int off = 16; off >= 1; off >>= 1) d += __shfl_xor(d, off, 32);
        if (lane == 0) { pos_acc += w * softplusf(-d); cnt_acc += w; }
    }
    s_wavesum[wave][lane*4 + 0] = acc.x;
    s_wavesum[wave][lane*4 + 1] = acc.y;
    s_wavesum[wave][lane*4 + 2] = acc.z;
    s_wavesum[wave][lane*4 + 3] = acc.w;
    if (lane == 0) { s_posw[wave] = pos_acc; s_cntw[wave] = cnt_acc; }
    __syncthreads();

    if (tid < DIM) {
        float s = 0.f, c = 0.f;
        #pragma unroll
        for (int wv = 0; wv < 8; ++wv) { s += s_wavesum[wv][tid]; c += s_cntw[wv]; }
        s_avg[tid] = s / c;   // X_rep_avg[b][tid]
    }
    if (tid == 0) {
        float p = 0.f, c = 0.f;
        #pragma unroll
        for (int wv = 0; wv < 8; ++wv) { p += s_posw[wv]; c += s_cntw[wv]; }
        part_pos[b] = p / c;  // per-row pos loss
    }
    __syncthreads();

    // ---------- Phase B: neg_prods via V_WMMA_F32_16X16X4_F32 (full fp32) ----------
    // A (16x4 f32): lanes 0-15 hold M, K=0..1 ; lanes 16-31 hold M, K=2..3
    // B (4x16 f32): avg broadcast over all 16 columns -> every D column = matvec
    const int half = (lane < 16) ? 0 : 2;       // K sub-offset for this half-wave
    const int m    = lane & 15;                  // matrix row handled by this lane
    for (int T = wave; T < NTILE; T += 8) {      // wave-uniform loop: EXEC stays full
        const int n  = T * 16 + m;
        const int nc = (n < NNEG) ? n : (NNEG - 1);   // pad tail tile (masked later)
        const float* rp = emb + (size_t)xn[nc] * DIM;
        v8f c = {0.f, 0.f, 0.f, 0.f, 0.f, 0.f, 0.f, 0.f};
        #pragma unroll 8
        for (int k = 0; k < 32; ++k) {
            const int e = 4*k + half;
            v2f a  = *(const v2f*)(rp + e);        // A[m][4k+half .. +1]
            v2f bb = *(const v2f*)(s_avg + e);     // B[half..half+1][*] = avg slice
            c = __builtin_amdgcn_wmma_f32_16x16x4_f32(
                    false, a, false, bb, (short)0, c, false, false);
        }
        // D layout: VGPR r = (M=r, lanes 0-15) / (M=r+8, lanes 16-31); col 0 suffices
        if (lane == 0) {
            #pragma unroll
            for (int r = 0; r < 8; ++r) s_negprod[T*16 + r] = c[r];
        } else if (lane == 16) {
            #pragma unroll
            for (int r = 0; r < 8; ++r) s_negprod[T*16 + 8 + r] = c[r];
        }
    }
    __syncthreads();

    // ---------- Phase C: top-50 by stable rank + neg loss ----------
    // hard_neg dot products ARE the top-50 neg_prods values (clip is a no-op),
    // so no re-gather / second matvec is needed.
    float contrib = 0.f;
    if (tid < NNEG) {
        const float v = s_negprod[tid];
        int rank = 0;
        for (int j = 0; j < NNEG; ++j) {
            const float vj = s_negprod[j];
            rank += (int)((vj > v) || (vj == v && j < tid));
        }
        const float mult = ((tid < NEASY) ? 1.f : 0.f) + ((rank < NHARD) ? 1.f : 0.f);
        contrib = mult * softplusf(v);
    }
    s_red[tid] = contrib;
    __syncthreads();
    #pragma unroll
    for (int st = 128; st >= 1; st >>= 1) {       // fixed-order deterministic tree
        if (tid < st) s_red[tid] += s_red[tid + st];
        __syncthreads();
    }
    if (tid == 0) part_neg[b] = s_red[0];
}

__global__ __launch_bounds__(256) void MFTF_reduce_kernel(
    const float* __restrict__ part_pos,
    const float* __restrict__ part_neg,
    float*       __restrict__ out)
{
    __shared__ float sp[256];
    __shared__ float sn[256];
    const int t = threadIdx.x;
    sp[t] = part_pos[t];
    sn[t] = part_neg[t];
    __syncthreads();
    #pragma unroll
    for (int st = 128; st >= 1; st >>= 1) {
        if (t < st) { sp[t] += sp[t + st]; sn[t] += sn[t + st]; }
        __syncthreads();
    }
    if (t == 0) out[0] = sp[0] * (1.0f / 256.0f)
                       + sn[0] * (1.0f / (256.0f * (NEASY + NHARD)));
}

extern "C" void kernel_launch(void* const* d_in, const int* in_sizes, int n_in,
                              void* d_out, int out_size, void* d_ws, size_t ws_size,
                              hipStream_t stream) {
    const int*   x_pos = (const int*)d_in[0];   // (256, 51)  int32
    const int*   x_neg = (const int*)d_in[1];   // (256, 200) int32
    const float* emb   = (const float*)d_in[2]; // (1000000, 128) f32

    float* part_pos = (float*)d_ws;             // 256 floats
    float* part_neg = part_pos + 256;           // 256 floats

    MFTF_main_kernel<<<256, 256, 0, stream>>>(x_pos, x_neg, emb, part_pos, part_neg);
    MFTF_reduce_kernel<<<1, 256, 0, stream>>>(part_pos, part_neg, (float*)d_out);
}